// PAM_Module_71073118814783
// MI455X (gfx1250) — compile-verified
//
#include <hip/hip_runtime.h>
#include <hip/hip_bf16.h>

// PAM module, fp32 end-to-end on CDNA5 WMMA (V_WMMA_F32_16X16X4_F32).
// Pipeline: proj_nd (Q), proj_nd (K^T), proj_cn (V), flash_attn (fused
// QK^T -> online softmax -> @V^T -> gamma*out + x). Energy matrix (268MB)
// is never materialized; workspace = Q(2MB) + K(2MB) + V(16MB) = ~21MB.

typedef __attribute__((ext_vector_type(2))) float v2f;
typedef __attribute__((ext_vector_type(8))) float v8f;

#define WMMA_F32(a, b, c) \
  __builtin_amdgcn_wmma_f32_16x16x4_f32(false, (a), false, (b), (short)0, (c), false, false)

static __device__ __forceinline__ v2f ld2(const float* p) {
  return *reinterpret_cast<const v2f*>(p);
}

// out[b][n][d] = sum_c W[d][c] * x[b][c][n] + bias[d]   (out layout B x N x D)
// A-frag = X^T tile (16 n x 4 c), B-frag = W^T tile (4 c x 16 d).
__global__ __launch_bounds__(32) void proj_nd(const float* __restrict__ x,
                                              const float* __restrict__ W,
                                              const float* __restrict__ bias,
                                              float* __restrict__ out,
                                              int Cin, int N, int D) {
  const int lane = threadIdx.x;
  const int half = lane >> 4, lr = lane & 15;
  const int n0 = blockIdx.x * 16, d0 = blockIdx.y * 16, b = blockIdx.z;
  const float* xb = x + (size_t)b * Cin * N;

  v8f acc = {0.f, 0.f, 0.f, 0.f, 0.f, 0.f, 0.f, 0.f};
  for (int k0 = 0; k0 < Cin; k0 += 4) {
    const int ks = k0 + 2 * half;  // lanes 16-31 carry K=2,3 of the 4-window
    v2f a;
    a.x = xb[(size_t)ks * N + n0 + lr];
    a.y = xb[(size_t)(ks + 1) * N + n0 + lr];
    v2f bf = ld2(W + (size_t)(d0 + lr) * Cin + ks);
    acc = WMMA_F32(a, bf, acc);
  }

  const float bd = bias[d0 + lr];
  float* ob = out + (size_t)b * N * D;
#pragma unroll
  for (int r = 0; r < 8; ++r) {
    const int n = n0 + r + 8 * half;  // C/D layout: VGPR r -> rows r / r+8
    ob[(size_t)n * D + d0 + lr] = acc[r] + bd;
  }
}

// out[b][c][n] = sum_k W[c][k] * x[b][k][n] + bias[c]   (out layout B x C x N)
// A-frag = W tile (16 c x 4 k), B-frag = X tile (4 k x 16 n).
__global__ __launch_bounds__(32) void proj_cn(const float* __restrict__ x,
                                              const float* __restrict__ W,
                                              const float* __restrict__ bias,
                                              float* __restrict__ out,
                                              int Cin, int N, int Cout) {
  const int lane = threadIdx.x;
  const int half = lane >> 4, lr = lane & 15;
  const int n0 = blockIdx.x * 16, c0 = blockIdx.y * 16, b = blockIdx.z;
  const float* xb = x + (size_t)b * Cin * N;

  v8f acc = {0.f, 0.f, 0.f, 0.f, 0.f, 0.f, 0.f, 0.f};
  for (int k0 = 0; k0 < Cin; k0 += 4) {
    const int ks = k0 + 2 * half;
    v2f a = ld2(W + (size_t)(c0 + lr) * Cin + ks);
    v2f bf;
    bf.x = xb[(size_t)ks * N + n0 + lr];
    bf.y = xb[(size_t)(ks + 1) * N + n0 + lr];
    acc = WMMA_F32(a, bf, acc);
  }

  float* ob = out + (size_t)b * Cout * N;
#pragma unroll
  for (int r = 0; r < 8; ++r) {
    const int c = c0 + r + 8 * half;
    ob[(size_t)c * N + n0 + lr] = acc[r] + bias[c];
  }
}

// Fused attention: one wave handles a 16-query tile x CT*16 output channels.
// Q: (B,N,D) ; Kt: (B,N,D) ; V: (B,C,N) ; out[b][c][m] = g*(P@V')[m][c] + x[b][c][m]
__global__ __launch_bounds__(32) void flash_attn(const float* __restrict__ Q,
                                                 const float* __restrict__ Kt,
                                                 const float* __restrict__ V,
                                                 const float* __restrict__ x,
                                                 const float* __restrict__ gamma,
                                                 float* __restrict__ out,
                                                 int N, int C, int D) {
  constexpr int CT = 8;  // 8 c-tiles of 16 -> 128 channels per wave
  const int lane = threadIdx.x;
  const int half = lane >> 4, lr = lane & 15;
  const int m0 = blockIdx.x * 16;
  const int cbase = blockIdx.y * (CT * 16);
  const int b = blockIdx.z;

  const float* Qb = Q + (size_t)b * N * D;
  const float* Kb = Kt + (size_t)b * N * D;
  const float* Vb = V + (size_t)b * C * N;

  // Q A-fragments: 8 K-steps of 4 over D=32 (reused for every key tile)
  v2f qa[8];
#pragma unroll
  for (int kk = 0; kk < 8; ++kk)
    qa[kk] = ld2(Qb + (size_t)(m0 + lr) * D + 4 * kk + 2 * half);

  float mrow[8], lsum[8];
#pragma unroll
  for (int r = 0; r < 8; ++r) {
    mrow[r] = -__builtin_inff();
    lsum[r] = 0.f;
  }
  v8f oacc[CT];
#pragma unroll
  for (int t = 0; t < CT; ++t) oacc[t] = (v8f){0.f, 0.f, 0.f, 0.f, 0.f, 0.f, 0.f, 0.f};

  __shared__ float Pl[16][17];  // prob tile bounce buffer (padded)

  for (int nt = 0; nt < N; nt += 16) {
    if (nt + 16 < N) __builtin_prefetch(Kb + (size_t)(nt + 16 + lr) * D, 0, 3);

    // S = Q * K^T (16x16), 8 x wmma over d
    v8f s = {0.f, 0.f, 0.f, 0.f, 0.f, 0.f, 0.f, 0.f};
#pragma unroll
    for (int kk = 0; kk < 8; ++kk) {
      v2f kb = ld2(Kb + (size_t)(nt + lr) * D + 4 * kk + 2 * half);
      s = WMMA_F32(qa[kk], kb, s);
    }

    // Online softmax; row M = r + 8*half lives across the 16 lanes of a half
#pragma unroll
    for (int r = 0; r < 8; ++r) {
      float sv = s[r];
      float rmax = sv;
      rmax = fmaxf(rmax, __shfl_xor(rmax, 1, 32));
      rmax = fmaxf(rmax, __shfl_xor(rmax, 2, 32));
      rmax = fmaxf(rmax, __shfl_xor(rmax, 4, 32));
      rmax = fmaxf(rmax, __shfl_xor(rmax, 8, 32));
      const float mnew = fmaxf(mrow[r], rmax);
      const float scale = __expf(mrow[r] - mnew);
      const float p = __expf(sv - mnew);
      float ps = p;
      ps += __shfl_xor(ps, 1, 32);
      ps += __shfl_xor(ps, 2, 32);
      ps += __shfl_xor(ps, 4, 32);
      ps += __shfl_xor(ps, 8, 32);
      lsum[r] = lsum[r] * scale + ps;
      mrow[r] = mnew;
#pragma unroll
      for (int t = 0; t < CT; ++t) oacc[t][r] *= scale;
      Pl[r + 8 * half][lr] = p;  // transpose bounce: D-layout -> LDS row-major
    }
    __syncthreads();

    // O += P * V'   (A-frag of P re-loaded from LDS in 16x4 A layout)
#pragma unroll
    for (int k0 = 0; k0 < 16; k0 += 4) {
      v2f pa;
      pa.x = Pl[lr][k0 + 2 * half];
      pa.y = Pl[lr][k0 + 2 * half + 1];
#pragma unroll
      for (int t = 0; t < CT; ++t) {
        const int c = cbase + t * 16 + lr;
        v2f vb = ld2(Vb + (size_t)c * N + nt + k0 + 2 * half);
        oacc[t] = WMMA_F32(pa, vb, oacc[t]);
      }
    }
    __syncthreads();
  }

  const float g = gamma[0];
#pragma unroll
  for (int r = 0; r < 8; ++r) {
    const float inv = 1.0f / lsum[r];
    const int m = m0 + r + 8 * half;
#pragma unroll
    for (int t = 0; t < CT; ++t) {
      const int c = cbase + t * 16 + lr;
      const size_t idx = ((size_t)b * C + c) * (size_t)N + m;
      out[idx] = g * (oacc[t][r] * inv) + x[idx];
    }
  }
}

extern "C" void kernel_launch(void* const* d_in, const int* in_sizes, int n_in,
                              void* d_out, int out_size, void* d_ws, size_t ws_size,
                              hipStream_t stream) {
  const float* x = (const float*)d_in[0];
  const float* Wq = (const float*)d_in[1];
  const float* bq = (const float*)d_in[2];
  const float* Wk = (const float*)d_in[3];
  const float* bk = (const float*)d_in[4];
  const float* Wv = (const float*)d_in[5];
  const float* bv = (const float*)d_in[6];
  const float* gamma = (const float*)d_in[7];
  float* out = (float*)d_out;

  constexpr int B = 4, C = 256, D = 32, N = 4096;

  float* Qws = (float*)d_ws;                 // B*N*D floats (2 MB)
  float* Kws = Qws + (size_t)B * N * D;      // B*N*D floats (2 MB)
  float* Vws = Kws + (size_t)B * N * D;      // B*C*N floats (16 MB)

  dim3 blk(32);
  proj_nd<<<dim3(N / 16, D / 16, B), blk, 0, stream>>>(x, Wq, bq, Qws, C, N, D);
  proj_nd<<<dim3(N / 16, D / 16, B), blk, 0, stream>>>(x, Wk, bk, Kws, C, N, D);
  proj_cn<<<dim3(N / 16, C / 16, B), blk, 0, stream>>>(x, Wv, bv, Vws, C, N, C);
  flash_attn<<<dim3(N / 16, C / 128, B), blk, 0, stream>>>(Qws, Kws, Vws, x, gamma, out, N, C, D);
}